// EmmaGAT_15152644620656
// MI455X (gfx1250) — compile-verified
//
#include <hip/hip_runtime.h>

#define NEG_SLOPE 0.2f
#define GAT_EPS 1e-16f
#define LN_EPS 1e-5f

typedef float v2f __attribute__((ext_vector_type(2)));
typedef float v8f __attribute__((ext_vector_type(8)));

// ---- monotonic float<->uint mapping for atomic float max ----
__device__ __forceinline__ unsigned mono_encode(float f) {
    unsigned b = __float_as_uint(f);
    return (b & 0x80000000u) ? ~b : (b | 0x80000000u);
}
__device__ __forceinline__ float mono_decode(unsigned u) {
    return (u & 0x80000000u) ? __uint_as_float(u ^ 0x80000000u)
                             : __uint_as_float(~u);
}
#define MONO_NEG_INF 0x007FFFFFu  // mono_encode(-inf)

// =====================================================================
// GEMM: Hout[n, OW] = X[n, K] @ W[OW, K]^T  using V_WMMA_F32_16X16X4_F32
// One wave32 computes one 16x16 output tile; K-loop in steps of 4.
// A tile (16x4 f32): lane m<16 holds row m, K=k0,k0+1 in v0,v1;
//                    lane 16+m holds row m, K=k0+2,k0+3.
// B tile (4x16 f32): lane n<16 holds col n (= W row), same K striping.
// C/D (16x16 f32, 8 VGPRs): vgpr r, lanes0-15 -> M=r, lanes16-31 -> M=r+8.
// =====================================================================
__global__ void gat_gemm_wmma(const float* __restrict__ X,
                              const float* __restrict__ W,
                              float* __restrict__ Hout,
                              int n, int K, int OW) {
    int wave = (int)((blockIdx.x * (long long)blockDim.x + threadIdx.x) >> 5);
    int lane = threadIdx.x & 31;
    int tilesN = OW >> 4;
    int tm = wave / tilesN;
    int tn = wave % tilesN;
    if (tm * 16 >= n) return;           // uniform per wave -> EXEC stays full

    int rowA = tm * 16 + (lane & 15);   // X row for A operand
    int rowB = tn * 16 + (lane & 15);   // W row (= output column) for B operand
    int khi  = (lane >> 4) * 2;         // 0 or 2 K-offset for upper half-wave

    const float* xp = X + (size_t)rowA * K + khi;
    const float* wp = W + (size_t)rowB * K + khi;

    v8f acc = {};
    for (int k = 0; k < K; k += 4) {
        v2f a, b;
        a.x = xp[k];     a.y = xp[k + 1];
        b.x = wp[k];     b.y = wp[k + 1];
        acc = __builtin_amdgcn_wmma_f32_16x16x4_f32(
            false, a, false, b, (short)0, acc, false, false);
    }

    int crow = tm * 16 + (lane >> 4) * 8;
    int ccol = tn * 16 + (lane & 15);
    float* op = Hout + (size_t)crow * OW + ccol;
#pragma unroll
    for (int r = 0; r < 8; ++r)
        op[(size_t)r * OW] = acc[r];
}

// =====================================================================
// Per (node, head): sa/da dots, init max/denom, zero agg slice.
// =====================================================================
__global__ void gat_node_prep(const float* __restrict__ H,
                              const float* __restrict__ att_s,
                              const float* __restrict__ att_d,
                              float* __restrict__ sa, float* __restrict__ da,
                              unsigned* __restrict__ m, float* __restrict__ den,
                              float* __restrict__ agg,
                              int n, int heads, int OW) {
    int t = blockIdx.x * blockDim.x + threadIdx.x;
    if (t >= n * heads) return;
    int node = t / heads;
    int hd   = t - node * heads;

    const float* hp = H + (size_t)node * OW + hd * 32;
    const float* as = att_s + hd * 32;
    const float* ad = att_d + hd * 32;
    float s = 0.f, d = 0.f;
#pragma unroll
    for (int c = 0; c < 32; ++c) {
        float v = hp[c];
        s += v * as[c];
        d += v * ad[c];
    }
    sa[t] = s;
    da[t] = d;
    m[t] = MONO_NEG_INF;
    den[t] = 0.f;
    float* ag = agg + (size_t)node * OW + hd * 32;
#pragma unroll
    for (int c = 0; c < 32; ++c) ag[c] = 0.f;
}

// =====================================================================
// Per edge: logits + atomic segment-max (monotonic uint atomicMax).
// =====================================================================
__global__ void gat_edge_logit_max(const float* __restrict__ sa,
                                   const float* __restrict__ da,
                                   const int* __restrict__ src,
                                   const int* __restrict__ dst,
                                   unsigned* __restrict__ m,
                                   float* __restrict__ wbuf,
                                   int E, int heads) {
    int e = blockIdx.x * blockDim.x + threadIdx.x;
    if (e >= E) return;
    int s = src[e], d = dst[e];
    for (int hd = 0; hd < heads; ++hd) {
        float lg = sa[(size_t)s * heads + hd] + da[(size_t)d * heads + hd];
        lg = (lg >= 0.f) ? lg : NEG_SLOPE * lg;
        wbuf[(size_t)e * heads + hd] = lg;
        atomicMax(&m[(size_t)d * heads + hd], mono_encode(lg));
    }
}

// =====================================================================
// Per (edge, 4-channel group): softmax weight + scatter-add messages.
// gshift = log2(OW/4) threads per edge.
// =====================================================================
__global__ void gat_edge_accum(const float* __restrict__ H,
                               const float* __restrict__ wbuf,
                               const unsigned* __restrict__ m,
                               const int* __restrict__ src,
                               const int* __restrict__ dst,
                               float* __restrict__ den,
                               float* __restrict__ agg,
                               int E, int heads, int OW, int gshift) {
    long long t = (long long)blockIdx.x * blockDim.x + threadIdx.x;
    int e = (int)(t >> gshift);
    if (e >= E) return;
    int j = (int)(t & ((1 << gshift) - 1));
    int c0 = j * 4;
    int hd = c0 >> 5;

    int s = src[e], d = dst[e];
    float mm = mono_decode(m[(size_t)d * heads + hd]);
    float w = __expf(wbuf[(size_t)e * heads + hd] - mm);
    if ((c0 & 31) == 0)
        atomicAdd(&den[(size_t)d * heads + hd], w);

    const float4 hs = *(const float4*)(H + (size_t)s * OW + c0);
    float* ag = agg + (size_t)d * OW + c0;
    atomicAdd(ag + 0, hs.x * w);
    atomicAdd(ag + 1, hs.y * w);
    atomicAdd(ag + 2, hs.z * w);
    atomicAdd(ag + 3, hs.w * w);
}

// =====================================================================
// Wave-per-node epilogue: agg/denom + residual + bias, optional LN+ReLU.
// OW = 64 (2 channels/lane) or 32 (1 channel/lane, no LN).
// =====================================================================
__global__ void gat_finalize(const float* __restrict__ H,
                             const float* __restrict__ agg,
                             const float* __restrict__ den,
                             const float* __restrict__ bias,
                             const float* __restrict__ gamma,
                             const float* __restrict__ beta,
                             float* __restrict__ out,
                             int n, int heads, int OW, int use_ln) {
    int node = (int)((blockIdx.x * (long long)blockDim.x + threadIdx.x) >> 5);
    int lane = threadIdx.x & 31;
    if (node >= n) return;

    float v0 = 0.f, v1 = 0.f;
    {
        int c = lane;
        float dn = fmaxf(den[(size_t)node * heads + (c >> 5)], GAT_EPS);
        v0 = agg[(size_t)node * OW + c] / dn + H[(size_t)node * OW + c] + bias[c];
    }
    if (OW == 64) {
        int c = lane + 32;
        float dn = fmaxf(den[(size_t)node * heads + (c >> 5)], GAT_EPS);
        v1 = agg[(size_t)node * OW + c] / dn + H[(size_t)node * OW + c] + bias[c];
    }

    if (use_ln) {
        float s = v0 + v1;
        for (int off = 16; off > 0; off >>= 1) s += __shfl_xor(s, off, 32);
        float mean = s * (1.0f / 64.0f);
        float d0 = v0 - mean, d1 = v1 - mean;
        float q = d0 * d0 + d1 * d1;
        for (int off = 16; off > 0; off >>= 1) q += __shfl_xor(q, off, 32);
        float rstd = rsqrtf(q * (1.0f / 64.0f) + LN_EPS);
        v0 = fmaxf(d0 * rstd * gamma[lane] + beta[lane], 0.f);
        v1 = fmaxf(d1 * rstd * gamma[lane + 32] + beta[lane + 32], 0.f);
    }

    out[(size_t)node * OW + lane] = v0;
    if (OW == 64) out[(size_t)node * OW + lane + 32] = v1;
}

// =====================================================================
// Host orchestration
// =====================================================================
static void run_layer(const float* xin, int K, const float* W,
                      const float* att_s, const float* att_d, const float* bias,
                      int heads, const float* gamma, const float* beta,
                      float* out, int use_ln,
                      int N, int E, const int* esrc, const int* edst,
                      float* hbuf, float* aggbuf, float* sabuf, float* dabuf,
                      unsigned* mbuf, float* denbuf, float* wbuf,
                      hipStream_t stream) {
    int OW = heads * 32;
    // GEMM
    {
        int waves = ((N + 15) / 16) * (OW / 16);
        long long threads = (long long)waves * 32;
        int blocks = (int)((threads + 255) / 256);
        gat_gemm_wmma<<<blocks, 256, 0, stream>>>(xin, W, hbuf, N, K, OW);
    }
    // node prep (sa/da, init m/den, zero agg)
    {
        int total = N * heads;
        gat_node_prep<<<(total + 255) / 256, 256, 0, stream>>>(
            hbuf, att_s, att_d, sabuf, dabuf, mbuf, denbuf, aggbuf, N, heads, OW);
    }
    // edge logits + segment max
    gat_edge_logit_max<<<(E + 255) / 256, 256, 0, stream>>>(
        sabuf, dabuf, esrc, edst, mbuf, wbuf, E, heads);
    // edge softmax + scatter-add
    {
        int gshift = (OW == 64) ? 4 : 3;
        long long total = (long long)E << gshift;
        int blocks = (int)((total + 255) / 256);
        gat_edge_accum<<<blocks, 256, 0, stream>>>(
            hbuf, wbuf, mbuf, esrc, edst, denbuf, aggbuf, E, heads, OW, gshift);
    }
    // epilogue
    {
        long long threads = (long long)N * 32;
        int blocks = (int)((threads + 255) / 256);
        gat_finalize<<<blocks, 256, 0, stream>>>(
            hbuf, aggbuf, denbuf, bias, gamma, beta, out, N, heads, OW, use_ln);
    }
}

extern "C" void kernel_launch(void* const* d_in, const int* in_sizes, int n_in,
                              void* d_out, int out_size, void* d_ws, size_t ws_size,
                              hipStream_t stream) {
    const float* x    = (const float*)d_in[0];
    const int*   esrc = (const int*)d_in[1];
    const int*   edst = (const int*)d_in[2];
    const float* W1   = (const float*)d_in[3];
    const float* as1  = (const float*)d_in[4];
    const float* ad1  = (const float*)d_in[5];
    const float* b1   = (const float*)d_in[6];
    const float* W2   = (const float*)d_in[7];
    const float* as2  = (const float*)d_in[8];
    const float* ad2  = (const float*)d_in[9];
    const float* b2   = (const float*)d_in[10];
    const float* W3   = (const float*)d_in[11];
    const float* as3  = (const float*)d_in[12];
    const float* ad3  = (const float*)d_in[13];
    const float* b3   = (const float*)d_in[14];
    const float* g1   = (const float*)d_in[15];
    const float* be1  = (const float*)d_in[16];
    const float* g2   = (const float*)d_in[17];
    const float* be2  = (const float*)d_in[18];

    int N = in_sizes[0] / 128;
    int E = in_sizes[1];

    // workspace layout (floats)
    float* ws    = (float*)d_ws;
    float* hbuf  = ws;                          // N*64
    float* aggb  = hbuf + (size_t)N * 64;       // N*64
    float* xcur  = aggb + (size_t)N * 64;       // N*64
    float* sab   = xcur + (size_t)N * 64;       // N*2
    float* dab   = sab + (size_t)N * 2;         // N*2
    unsigned* mb = (unsigned*)(dab + (size_t)N * 2);   // N*2
    float* denb  = (float*)mb + (size_t)N * 2;  // N*2
    float* wbuf  = denb + (size_t)N * 2;        // E*2

    // layer 1: 128 -> 64 (H=2), LN+ReLU -> xcur
    run_layer(x, 128, W1, as1, ad1, b1, 2, g1, be1, xcur, 1,
              N, E, esrc, edst, hbuf, aggb, sab, dab, mb, denb, wbuf, stream);
    // layer 2: 64 -> 64 (H=2), LN+ReLU -> xcur
    run_layer(xcur, 64, W2, as2, ad2, b2, 2, g2, be2, xcur, 1,
              N, E, esrc, edst, hbuf, aggb, sab, dab, mb, denb, wbuf, stream);
    // layer 3: 64 -> 32 (H=1), no LN -> d_out
    run_layer(xcur, 64, W3, as3, ad3, b3, 1, nullptr, nullptr,
              (float*)d_out, 0,
              N, E, esrc, edst, hbuf, aggb, sab, dab, mb, denb, wbuf, stream);
}